// SnakeBetaAA_72516227826168
// MI455X (gfx1250) — compile-verified
//
#include <hip/hip_runtime.h>

typedef __attribute__((ext_vector_type(2))) float v2f;
typedef __attribute__((ext_vector_type(8))) float v8f;

// Problem constants (match reference)
constexpr int kB    = 8;
constexpr int kC    = 512;
constexpr int kT    = 32768;
constexpr int kOutT = kT + 1;        // even kernel -> T+1 outputs
constexpr int kKS   = 12;
constexpr int kPAD  = 6;             // KERNEL_SIZE / 2
constexpr float kEPS = 1e-9f;

// Tiling
constexpr int kCG     = 16;               // channels per workgroup tile
constexpr int kWAVES  = 8;                // 256 threads, wave32
constexpr int kCHUNK  = kWAVES * 16;      // 128 output times per WG
constexpr int kLROW   = kCHUNK + kKS;     // 140 input samples per channel needed
constexpr int kLPITCH = 146;              // even pitch; bank stride 18 -> conflict-free

__global__ __launch_bounds__(256)
void snakebeta_aa_wmma(const float* __restrict__ x,
                       const float* __restrict__ alpha,
                       const float* __restrict__ beta,
                       const float* __restrict__ filt,
                       float* __restrict__ out)
{
    __shared__ float lds[kCG * kLPITCH];   // 9344 B

    const int chunk = blockIdx.x;          // time chunk
    const int c0    = blockIdx.y * kCG;    // channel group base
    const int bb    = blockIdx.z;          // batch
    const int t0    = chunk * kCHUNK;      // first output time of this WG

    const int tid  = threadIdx.x;
    const int lane = tid & 31;
    const int wv   = tid >> 5;             // wave id 0..7
    const int half = lane >> 4;            // 0: lanes 0-15, 1: lanes 16-31
    const int lm   = lane & 15;

    // ---- B fragments: banded Toeplitz filter matrix, 7 K-groups of 4 ----
    // B[s][n] = h[s - n] for 0 <= s-n < 12, n = output-time lane (lm).
    // Layout guess (symmetric w/ A): VGPR0 = K rows {0,2}+4g by half, VGPR1 = +1.
    v2f bf[7];
#pragma unroll
    for (int g = 0; g < 7; ++g) {
        const int s0 = 4 * g + 2 * half;
        const int j0 = s0 - lm;
        const int j1 = j0 + 1;
        bf[g].x = (j0 >= 0 && j0 < kKS) ? filt[j0] : 0.0f;
        bf[g].y = (j1 >= 0 && j1 < kKS) ? filt[j1] : 0.0f;
    }

    // ---- cooperative load + fused SnakeBeta into LDS (reflect padding) ----
    {
        const int ch = tid >> 4;                 // 0..15 channel within group
        const int tl = tid & 15;                 // 16 consecutive t per channel
        const int c  = c0 + ch;
        const float ea     = __expf(alpha[c]);
        const float inv_eb = 1.0f / (__expf(beta[c]) + kEPS);
        const float* __restrict__ xrow = x + (size_t)(bb * kC + c) * kT;
        float* __restrict__ lrow = lds + ch * kLPITCH;
#pragma unroll
        for (int j = 0; j < 9; ++j) {
            const int s = tl + 16 * j;
            if (s < kLROW) {
                int i = t0 - kPAD + s;
                if (i < 0)   i = -i;                  // reflect left
                if (i >= kT) i = 2 * kT - 2 - i;      // reflect right
                const float v  = xrow[i];
                const float sn = __sinf(v * ea);
                lrow[s] = v + sn * sn * inv_eb;       // SnakeBeta
            }
        }
    }
    __syncthreads();

    // ---- per-wave 16(ch) x 16(time) tile: 7x V_WMMA_F32_16X16X4_F32 ----
    // A[m][k] = snake(x)[ch m][sbase + 4g + k] from LDS.
    // A layout: lanes 0-15 -> M=lane, VGPRs {K0,K1}; lanes 16-31 -> {K2,K3}.
    const int sbase = wv * 16;
    const float* __restrict__ arow = lds + lm * kLPITCH + sbase + 2 * half;
    v8f acc = {0.f, 0.f, 0.f, 0.f, 0.f, 0.f, 0.f, 0.f};
#pragma unroll
    for (int g = 0; g < 7; ++g) {
        const v2f af = *(const v2f*)(arow + 4 * g);   // 8B-aligned ds_load_b64
        acc = __builtin_amdgcn_wmma_f32_16x16x4_f32(
            /*neg_a=*/false, af, /*neg_b=*/false, bf[g],
            /*c_mod=*/(short)0, acc, /*reuse_a=*/false, /*reuse_b=*/false);
    }

    // ---- coalesced stores: lane = time (N), VGPR index = channel (M) ----
    const int t = t0 + sbase + lm;
    if (t < kOutT) {
#pragma unroll
        for (int v = 0; v < 8; ++v) {
            const int m = v + 8 * half;                // channel within group
            out[(size_t)(bb * kC + c0 + m) * kOutT + t] = acc[v];
        }
    }
}

extern "C" void kernel_launch(void* const* d_in, const int* in_sizes, int n_in,
                              void* d_out, int out_size, void* d_ws, size_t ws_size,
                              hipStream_t stream) {
    (void)in_sizes; (void)n_in; (void)out_size; (void)d_ws; (void)ws_size;
    const float* x     = (const float*)d_in[0];
    const float* alpha = (const float*)d_in[1];
    const float* beta  = (const float*)d_in[2];
    const float* filt  = (const float*)d_in[3];
    float* out = (float*)d_out;

    dim3 grid((kOutT + kCHUNK - 1) / kCHUNK,  // 257 time chunks
              kC / kCG,                       // 32 channel groups
              kB);                            // 8 batches
    snakebeta_aa_wmma<<<grid, 256, 0, stream>>>(x, alpha, beta, filt, out);
}